// GIN_50723563766315
// MI455X (gfx1250) — compile-verified
//
#include <hip/hip_runtime.h>

// ---------------------------------------------------------------------------
// GIN forward (3 layers) for MI455X / gfx1250, fp32 end-to-end.
// Matmuls use V_WMMA_F32_16X16X4_F32 (full-precision f32 WMMA).
// ---------------------------------------------------------------------------

typedef __attribute__((ext_vector_type(2))) float v2f;
typedef __attribute__((ext_vector_type(8))) float v8f;

#define N_NODES 50000
#define N_EDGES 1600000
#define HIDDEN  128
#define OUT_SZ  64
#define BN_EPS  1e-5f

__device__ __forceinline__ v8f wmma_f32x4(v2f a, v2f b, v8f c) {
    // 8 args: (neg_a, A, neg_b, B, c_mod, C, reuse_a, reuse_b)
    return __builtin_amdgcn_wmma_f32_16x16x4_f32(
        false, a, false, b, (short)0, c, false, false);
}

// x0 = (1 + eps[layer]) * h   (float4 vectorized; scatter adds on top of this)
__global__ void gin_init_x0(const float* __restrict__ h,
                            const float* __restrict__ eps, int layer,
                            float* __restrict__ x0) {
    int i = blockIdx.x * blockDim.x + threadIdx.x;
    float s = 1.0f + eps[layer];
    float4 v = ((const float4*)h)[i];
    ((float4*)x0)[i] = make_float4(v.x * s, v.y * s, v.z * s, v.w * s);
}

// One wave per edge: 32 lanes x float4 = 128 floats. Coalesced 512B gather of
// h[src], f32 atomic scatter into x0[dst] (resolves in L2; x0 is L2-resident).
__global__ void gin_scatter(const float* __restrict__ h,
                            const int* __restrict__ src,
                            const int* __restrict__ dst,
                            float* __restrict__ x0) {
    int gid = blockIdx.x * blockDim.x + threadIdx.x;
    int e = gid >> 5;
    int c = (gid & 31) * 4;
    int s = src[e];
    int d = dst[e];
    float4 v = *(const float4*)(h + (size_t)s * HIDDEN + c);
    float* p = x0 + (size_t)d * HIDDEN + c;
    atomicAdd(p + 0, v.x);
    atomicAdd(p + 1, v.y);
    atomicAdd(p + 2, v.z);
    atomicAdd(p + 3, v.w);
}

// X1[M,128] = X0[M,128] @ W[128,128] + bias.  One wave per 16x16 tile,
// 8 waves/block cover all 128 output columns of one 16-row stripe.
__global__ void gin_gemm1(const float* __restrict__ A,
                          const float* __restrict__ W,
                          const float* __restrict__ bias,
                          float* __restrict__ C) {
    const int lane = threadIdx.x & 31;
    const int wave = threadIdx.x >> 5;            // ntile 0..7
    const int m0   = blockIdx.x * 16;
    const int n    = wave * 16 + (lane & 15);
    const int kh   = (lane >> 4) * 2;             // 0 or 2
    const float* arow = A + (size_t)(m0 + (lane & 15)) * HIDDEN;
    v8f acc = {};
    #pragma unroll
    for (int k = 0; k < HIDDEN; k += 4) {
        v2f a; a.x = arow[k + kh];              a.y = arow[k + kh + 1];
        v2f b; b.x = W[(k + kh) * HIDDEN + n];  b.y = W[(k + kh + 1) * HIDDEN + n];
        acc = wmma_f32x4(a, b, acc);
    }
    const float bn = bias[n];
    const int mr = m0 + (lane >> 4) * 8;          // D: VGPR i -> row i (+8 hi half)
    #pragma unroll
    for (int i = 0; i < 8; ++i)
        C[(size_t)(mr + i) * HIDDEN + n] = acc[i] + bn;
}

__global__ void gin_stats_zero(float* __restrict__ colsum,
                               float* __restrict__ colsq) {
    int t = threadIdx.x;
    colsum[t] = 0.0f;
    colsq[t]  = 0.0f;
}

// Per-column sum / sum-of-squares over 50000 rows; thread t owns column t
// (coalesced row-major reads), one pair of atomics per block per column.
__global__ void gin_stats(const float* __restrict__ X,
                          float* __restrict__ colsum,
                          float* __restrict__ colsq) {
    int t = threadIdx.x;                          // column 0..127
    float s = 0.0f, q = 0.0f;
    for (int r = blockIdx.x; r < N_NODES; r += gridDim.x) {
        float v = X[(size_t)r * HIDDEN + t];
        s += v;
        q += v * v;
    }
    atomicAdd(&colsum[t], s);
    atomicAdd(&colsq[t],  q);
}

// Fold BN into per-K affine:  bn(x)_k = x * scale[k] + shift[k]
__global__ void gin_bn_finalize(const float* __restrict__ colsum,
                                const float* __restrict__ colsq,
                                const float* __restrict__ gamma,
                                const float* __restrict__ beta,
                                int layer,
                                float* __restrict__ scale,
                                float* __restrict__ shift) {
    int t = threadIdx.x;
    const float inv_n = 1.0f / (float)N_NODES;
    float mu  = colsum[t] * inv_n;
    float var = colsq[t] * inv_n - mu * mu;
    float sc  = gamma[layer * HIDDEN + t] * rsqrtf(var + BN_EPS);
    scale[t] = sc;
    shift[t] = beta[layer * HIDDEN + t] - mu * sc;
}

// H[M,128] = relu( relu(bn(X1)) @ W2 + b2 ); BN+ReLU fused into the A-load.
__global__ void gin_gemm2(const float* __restrict__ X,
                          const float* __restrict__ scale,
                          const float* __restrict__ shift,
                          const float* __restrict__ W,
                          const float* __restrict__ bias,
                          float* __restrict__ H) {
    const int lane = threadIdx.x & 31;
    const int wave = threadIdx.x >> 5;
    const int m0   = blockIdx.x * 16;
    const int n    = wave * 16 + (lane & 15);
    const int kh   = (lane >> 4) * 2;
    const float* arow = X + (size_t)(m0 + (lane & 15)) * HIDDEN;
    v8f acc = {};
    #pragma unroll
    for (int k = 0; k < HIDDEN; k += 4) {
        int k0 = k + kh, k1 = k + kh + 1;
        v2f a;
        a.x = fmaxf(arow[k0] * scale[k0] + shift[k0], 0.0f);
        a.y = fmaxf(arow[k1] * scale[k1] + shift[k1], 0.0f);
        v2f b; b.x = W[k0 * HIDDEN + n]; b.y = W[k1 * HIDDEN + n];
        acc = wmma_f32x4(a, b, acc);
    }
    const float bn = bias[n];
    const int mr = m0 + (lane >> 4) * 8;
    #pragma unroll
    for (int i = 0; i < 8; ++i)
        H[(size_t)(mr + i) * HIDDEN + n] = fmaxf(acc[i] + bn, 0.0f);
}

// out[M,64] = H[M,128] @ Wout[128,64] + bout.  4 waves/block (N=64).
__global__ void gin_gemm_out(const float* __restrict__ H,
                             const float* __restrict__ Wout,
                             const float* __restrict__ bout,
                             float* __restrict__ out) {
    const int lane = threadIdx.x & 31;
    const int wave = threadIdx.x >> 5;            // ntile 0..3
    const int m0   = blockIdx.x * 16;
    const int n    = wave * 16 + (lane & 15);
    const int kh   = (lane >> 4) * 2;
    const float* arow = H + (size_t)(m0 + (lane & 15)) * HIDDEN;
    v8f acc = {};
    #pragma unroll
    for (int k = 0; k < HIDDEN; k += 4) {
        v2f a; a.x = arow[k + kh];             a.y = arow[k + kh + 1];
        v2f b; b.x = Wout[(k + kh) * OUT_SZ + n]; b.y = Wout[(k + kh + 1) * OUT_SZ + n];
        acc = wmma_f32x4(a, b, acc);
    }
    const float bn = bout[n];
    const int mr = m0 + (lane >> 4) * 8;
    #pragma unroll
    for (int i = 0; i < 8; ++i)
        out[(size_t)(mr + i) * OUT_SZ + n] = acc[i] + bn;
}

extern "C" void kernel_launch(void* const* d_in, const int* in_sizes, int n_in,
                              void* d_out, int out_size, void* d_ws, size_t ws_size,
                              hipStream_t stream) {
    const float* node_feat = (const float*)d_in[0];
    const int*   src       = (const int*)  d_in[1];
    const int*   dst       = (const int*)  d_in[2];
    const float* W1        = (const float*)d_in[3];
    const float* b1        = (const float*)d_in[4];
    const float* gamma     = (const float*)d_in[5];
    const float* beta      = (const float*)d_in[6];
    const float* W2        = (const float*)d_in[7];
    const float* b2        = (const float*)d_in[8];
    const float* eps       = (const float*)d_in[9];
    const float* Wout      = (const float*)d_in[10];
    const float* bout      = (const float*)d_in[11];

    const size_t NH = (size_t)N_NODES * HIDDEN;   // 6.4M floats
    float* ws     = (float*)d_ws;
    float* hA     = ws;                           // layer outputs (ping)
    float* hB     = hA + NH;                      // layer outputs (pong)
    float* X0     = hB + NH;                      // (1+eps)h + agg
    float* X1     = X0 + NH;                      // post-GEMM1
    float* colsum = X1 + NH;
    float* colsq  = colsum + HIDDEN;
    float* scale  = colsq  + HIDDEN;
    float* shift  = scale  + HIDDEN;

    const int mtiles = N_NODES / 16;              // 3125 exactly
    const float* hin = node_feat;
    float* houts[3] = {hA, hB, hA};

    for (int layer = 0; layer < 3; ++layer) {
        float* hout = houts[layer];
        gin_init_x0<<<(int)(NH / (256 * 4)), 256, 0, stream>>>(hin, eps, layer, X0);
        gin_scatter<<<N_EDGES * 32 / 256, 256, 0, stream>>>(hin, src, dst, X0);
        gin_gemm1<<<mtiles, 256, 0, stream>>>(X0, W1 + (size_t)layer * HIDDEN * HIDDEN,
                                              b1 + layer * HIDDEN, X1);
        gin_stats_zero<<<1, HIDDEN, 0, stream>>>(colsum, colsq);
        gin_stats<<<500, HIDDEN, 0, stream>>>(X1, colsum, colsq);
        gin_bn_finalize<<<1, HIDDEN, 0, stream>>>(colsum, colsq, gamma, beta, layer,
                                                  scale, shift);
        gin_gemm2<<<mtiles, 256, 0, stream>>>(X1, scale, shift,
                                              W2 + (size_t)layer * HIDDEN * HIDDEN,
                                              b2 + layer * HIDDEN, hout);
        hin = hout;
    }
    gin_gemm_out<<<mtiles, 128, 0, stream>>>(hin, Wout, bout, (float*)d_out);
}